// MaskEncoder_3393024164037
// MI455X (gfx1250) — compile-verified
//
#include <hip/hip_runtime.h>

typedef __attribute__((ext_vector_type(2))) float v2f;
typedef __attribute__((ext_vector_type(8))) float v8f;

#define NN 50000
#define NE 800000
#define IN_DIM 128
#define H_DIM 64
#define OUT_DIM 16
#define K_HOMO 640000u
#define K_HETERO 160000u

// ---------------- degree / normalization ----------------
__global__ void deg_init(float* deg) {
  int i = blockIdx.x * blockDim.x + threadIdx.x;
  if (i < NN) deg[i] = 1.0f;  // self-loop contributes 1
}
__global__ void deg_count(const int* __restrict__ dst, float* deg) {
  int e = blockIdx.x * blockDim.x + threadIdx.x;
  if (e < NE) atomicAdd(&deg[dst[e]], 1.0f);
}
__global__ void deg_rsqrt(float* deg) {
  int i = blockIdx.x * blockDim.x + threadIdx.x;
  if (i < NN) deg[i] = rsqrtf(deg[i]);  // deg >= 1 always
}

// ---------------- fp32 WMMA GEMM: C[M,N] = A[M,K] @ B[K,N] ----------------
// One 16x16 output tile per wave32 via V_WMMA_F32_16X16X4_F32, K stepped by 4.
template <int K, int N>
__global__ void gemm_wmma(const float* __restrict__ A, const float* __restrict__ B,
                          float* __restrict__ C, int mTiles) {
  int wid  = (int)((blockIdx.x * blockDim.x + threadIdx.x) >> 5);
  constexpr int NT = N / 16;
  int tm = wid / NT, tn = wid % NT;
  if (tm >= mTiles) return;                 // wave-uniform: EXEC stays all-ones
  int lane = threadIdx.x & 31;
  int half = lane >> 4;                     // 0: K pair {0,1}; 1: K pair {2,3}
  int l16  = lane & 15;
  const float* arow = A + (size_t)(tm * 16 + l16) * K;  // A: M row per lane
  const float* bcol = B + (size_t)(tn * 16 + l16);      // B: N col per lane
  v8f acc = {};
  for (int k0 = 0; k0 < K; k0 += 4) {
    int ka = k0 + half * 2;
    v2f a, b;
    a.x = arow[ka];
    a.y = arow[ka + 1];
    b.x = bcol[(size_t)ka * N];
    b.y = bcol[(size_t)(ka + 1) * N];
    acc = __builtin_amdgcn_wmma_f32_16x16x4_f32(false, a, false, b, (short)0,
                                                acc, false, false);
  }
  int n  = tn * 16 + l16;
  int mb = tm * 16 + half * 8;              // lanes 0-15: M=r, lanes 16-31: M=8+r
#pragma unroll
  for (int r = 0; r < 8; ++r) C[(size_t)(mb + r) * N + n] = acc[r];
}

// ---------------- GCN aggregation ----------------
template <int HD>
__global__ void self_loop(const float* __restrict__ h, const float* __restrict__ dinv,
                          float* __restrict__ agg) {
  unsigned i = blockIdx.x * blockDim.x + threadIdx.x;
  if (i >= (unsigned)NN * HD) return;
  unsigned node = i / HD;
  float di = dinv[node];
  agg[i] = di * di * h[i];
}

template <int HD>
__global__ void scatter_edges(const float* __restrict__ h, const float* __restrict__ dinv,
                              const int* __restrict__ src, const int* __restrict__ dst,
                              float* agg) {
  unsigned i = blockIdx.x * blockDim.x + threadIdx.x;
  if (i >= (unsigned)NE * HD) return;
  unsigned e = i / HD, f = i % HD;          // HD power-of-2 -> shifts/masks
  int s = src[e], d = dst[e];
  float nrm = dinv[s] * dinv[d];
  atomicAdd(&agg[(size_t)d * HD + f], nrm * h[(size_t)s * HD + f]);
}

template <int HD, bool RELU>
__global__ void finalize(float* agg, const float* __restrict__ bias) {
  unsigned i = blockIdx.x * blockDim.x + threadIdx.x;
  if (i >= (unsigned)NN * HD) return;
  float v = agg[i] + bias[i % HD];
  agg[i] = RELU ? fmaxf(v, 0.0f) : v;
}

// ---------------- edge scoring -> order-preserving uint keys ----------------
__global__ void edge_score(const float* __restrict__ xM2, const int* __restrict__ src,
                           const int* __restrict__ dst, unsigned* __restrict__ keys) {
  unsigned e = blockIdx.x * blockDim.x + threadIdx.x;
  if (e >= NE) return;
  const float4* a = (const float4*)(xM2 + (size_t)src[e] * OUT_DIM);
  const float4* b = (const float4*)(xM2 + (size_t)dst[e] * OUT_DIM);
  float v = 0.0f;
#pragma unroll
  for (int i = 0; i < 4; ++i) {
    float4 x = a[i], y = b[i];
    v += x.x * y.x + x.y * y.y + x.z * y.z + x.w * y.w;
  }
  unsigned u = __float_as_uint(v);
  keys[e] = (u & 0x80000000u) ? ~u : (u | 0x80000000u);  // larger float -> larger key
}

// ---------------- 4-pass MSB radix select for k-th largest ----------------
// state: [0]=prefix  [1]=krem  [2]=homoCnt  [3]=heteroCnt  [4]=eqCnt
__global__ void radix_init(unsigned* state, unsigned* hist) {
  if (threadIdx.x == 0) {
    state[0] = 0; state[1] = K_HOMO; state[2] = 0; state[3] = 0; state[4] = 0;
  }
  hist[threadIdx.x] = 0;  // 256 threads
}

__global__ void hist_pass(const unsigned* __restrict__ keys, unsigned* hist,
                          const unsigned* __restrict__ state, int shift, unsigned maskHigh) {
  __shared__ unsigned lh[256];
  lh[threadIdx.x] = 0;
  __syncthreads();
  unsigned prefix = state[0];
  unsigned e = blockIdx.x * blockDim.x + threadIdx.x;
  if (e < NE) {
    unsigned k = keys[e];
    if ((k & maskHigh) == prefix) atomicAdd(&lh[(k >> shift) & 0xFF], 1u);
  }
  __syncthreads();
  if (lh[threadIdx.x]) atomicAdd(&hist[threadIdx.x], lh[threadIdx.x]);
}

__global__ void select_pass(unsigned* hist, unsigned* state, int shift) {
  if (threadIdx.x == 0) {
    unsigned krem = state[1], acc = 0;
    int bin = 0;
    for (int b = 255; b >= 0; --b) {
      unsigned c = hist[b];
      if (acc + c >= krem) { bin = b; state[1] = krem - acc; break; }
      acc += c;
    }
    state[0] |= ((unsigned)bin) << shift;
  }
  __syncthreads();
  hist[threadIdx.x] = 0;  // reset for next pass (256 threads)
}

// ---------------- partition edges around threshold ----------------
__global__ void partition_edges(const unsigned* __restrict__ keys,
                                const int* __restrict__ src, const int* __restrict__ dst,
                                const unsigned* __restrict__ state, unsigned* cnt,
                                int* __restrict__ out) {
  unsigned e = blockIdx.x * blockDim.x + threadIdx.x;
  if (e >= NE) return;
  unsigned T = state[0], krem = state[1], k = keys[e];
  bool homo;
  if (k > T)      homo = true;
  else if (k < T) homo = false;
  else            homo = (atomicAdd(&cnt[2], 1u) < krem);  // tie-break on threshold
  if (homo) {
    unsigned s = atomicAdd(&cnt[0], 1u);
    out[s] = src[e];
    out[K_HOMO + s] = dst[e];
  } else {
    unsigned s = atomicAdd(&cnt[1], 1u);
    out[2 * K_HOMO + s] = src[e];
    out[2 * K_HOMO + K_HETERO + s] = dst[e];
  }
}

extern "C" void kernel_launch(void* const* d_in, const int* in_sizes, int n_in,
                              void* d_out, int out_size, void* d_ws, size_t ws_size,
                              hipStream_t stream) {
  const float* x  = (const float*)d_in[0];
  const int*   ei = (const int*)d_in[1];
  const float* W1 = (const float*)d_in[2];
  const float* b1 = (const float*)d_in[3];
  const float* W2 = (const float*)d_in[4];
  const float* b2 = (const float*)d_in[5];
  const int* src = ei;            // edge_index[0]
  const int* dst = ei + NE;       // edge_index[1]
  int* out = (int*)d_out;

  // workspace layout (floats/uints), 128-element aligned
  float* dinv = (float*)d_ws;                          // 50000
  float* h1   = dinv + 50176;                          // 3,200,000
  float* agg1 = h1 + (size_t)NN * H_DIM;               // 3,200,000 (becomes xM1)
  float* h2   = agg1 + (size_t)NN * H_DIM;             // 800,000
  float* agg2 = h2 + (size_t)NN * OUT_DIM;             // 800,000 (becomes xM2)
  unsigned* keys  = (unsigned*)(agg2 + (size_t)NN * OUT_DIM);  // 800,000
  unsigned* hist  = keys + NE;                         // 256
  unsigned* state = hist + 256;                        // 8

  const int B = 256;
  // normalization: dinv = rsqrt(1 + in-degree)
  deg_init<<<(NN + B - 1) / B, B, 0, stream>>>(dinv);
  deg_count<<<(NE + B - 1) / B, B, 0, stream>>>(dst, dinv);
  deg_rsqrt<<<(NN + B - 1) / B, B, 0, stream>>>(dinv);

  // layer 1: h1 = x @ W1 (WMMA), aggregate, +b1, relu
  {
    int mTiles = NN / 16, waves = mTiles * (H_DIM / 16);
    gemm_wmma<IN_DIM, H_DIM><<<(waves + 7) / 8, B, 0, stream>>>(x, W1, h1, mTiles);
  }
  self_loop<H_DIM><<<(NN * H_DIM + B - 1) / B, B, 0, stream>>>(h1, dinv, agg1);
  scatter_edges<H_DIM><<<(NE * H_DIM + B - 1) / B, B, 0, stream>>>(h1, dinv, src, dst, agg1);
  finalize<H_DIM, true><<<(NN * H_DIM + B - 1) / B, B, 0, stream>>>(agg1, b1);

  // layer 2: h2 = xM1 @ W2 (WMMA), aggregate, +b2
  {
    int mTiles = NN / 16, waves = mTiles * (OUT_DIM / 16);
    gemm_wmma<H_DIM, OUT_DIM><<<(waves + 7) / 8, B, 0, stream>>>(agg1, W2, h2, mTiles);
  }
  self_loop<OUT_DIM><<<(NN * OUT_DIM + B - 1) / B, B, 0, stream>>>(h2, dinv, agg2);
  scatter_edges<OUT_DIM><<<(NE * OUT_DIM + B - 1) / B, B, 0, stream>>>(h2, dinv, src, dst, agg2);
  finalize<OUT_DIM, false><<<(NN * OUT_DIM + B - 1) / B, B, 0, stream>>>(agg2, b2);

  // edge scores -> sortable keys
  edge_score<<<(NE + B - 1) / B, B, 0, stream>>>(agg2, src, dst, keys);

  // radix-select the K_HOMO-th largest key (threshold T, plus tie quota)
  radix_init<<<1, 256, 0, stream>>>(state, hist);
  const int      shifts[4] = {24, 16, 8, 0};
  const unsigned masks[4]  = {0x00000000u, 0xFF000000u, 0xFFFF0000u, 0xFFFFFF00u};
  for (int p = 0; p < 4; ++p) {
    hist_pass<<<(NE + B - 1) / B, B, 0, stream>>>(keys, hist, state, shifts[p], masks[p]);
    select_pass<<<1, 256, 0, stream>>>(hist, state, shifts[p]);
  }

  // partition: top 640000 -> homo block, bottom 160000 -> hetero block
  partition_edges<<<(NE + B - 1) / B, B, 0, stream>>>(keys, src, dst, state, state + 2, out);
}